// GConvGRUWrapper_21680994910528
// MI455X (gfx1250) — compile-verified
//
#include <hip/hip_runtime.h>
#include <hip/hip_bf16.h>

// ---------------------------------------------------------------------------
// GConvGRU with H0 = 0  =>  out = (1 - sigmoid(gx0+bh0)) * tanh(gx2+bh2) @ Wlin + blin
// where gx_g = [x | Tx] @ [Wx0_g ; Wx1_g] + bx_g and Tx = scatter(norm * x[src])
// ---------------------------------------------------------------------------

typedef float v2f __attribute__((ext_vector_type(2)));
typedef float v8f __attribute__((ext_vector_type(8)));

#define N_NODES 50000
#define N_EDGES 1600000
#define N_TILES 3125          // 50000 / 16, exact
#define TILES_PER_BLOCK 8     // 8 waves * 16 rows = 128 rows / block

// ---------------- small utility kernels ----------------

__global__ void zero_f32_kernel(float* __restrict__ p, int n) {
  int i = blockIdx.x * blockDim.x + threadIdx.x;
  if (i < n) p[i] = 0.0f;
}

__global__ void deg_kernel(const float* __restrict__ w,
                           const int* __restrict__ row,
                           float* __restrict__ deg, int E) {
  int e = blockIdx.x * blockDim.x + threadIdx.x;
  if (e < E) unsafeAtomicAdd(&deg[row[e]], w[e]);
}

__global__ void disq_kernel(const float* __restrict__ deg,
                            float* __restrict__ disq, int n) {
  int i = blockIdx.x * blockDim.x + threadIdx.x;
  if (i < n) {
    float d = deg[i];
    disq[i] = (d > 0.0f) ? rsqrtf(d) : 0.0f;
  }
}

__global__ void norm_kernel(const float* __restrict__ w,
                            const int* __restrict__ row,
                            const int* __restrict__ col,
                            const float* __restrict__ disq,
                            float* __restrict__ nrm, int E) {
  int e = blockIdx.x * blockDim.x + threadIdx.x;
  if (e < E) nrm[e] = -w[e] * disq[row[e]] * disq[col[e]];
}

// ---------------- edge scatter: Tx[col] += norm * x[row] ----------------
// One wave handles 32 edges per step: coalesced metadata load, then for each
// edge all 32 lanes (= 32 features) do a coalesced gather + f32 atomic add.

__global__ void scatter_kernel(const float* __restrict__ x,
                               const int* __restrict__ row,
                               const int* __restrict__ col,
                               const float* __restrict__ nrm,
                               float* __restrict__ Tx, int E) {
  int lane = threadIdx.x & 31;
  int gwave = (blockIdx.x * blockDim.x + threadIdx.x) >> 5;
  int nwaves = (gridDim.x * blockDim.x) >> 5;

  for (int e0 = gwave * 32; e0 < E; e0 += nwaves * 32) {
    int my = e0 + lane;                 // E is a multiple of 32
    int r = row[my];
    int c = col[my];
    float nm = nrm[my];
    #pragma unroll 4
    for (int j = 0; j < 32; ++j) {
      int   rj = __shfl(r, j, 32);
      int   cj = __shfl(c, j, 32);
      float nj = __shfl(nm, j, 32);
      float xv = x[rj * 32 + lane];
      unsafeAtomicAdd(&Tx[cj * 32 + lane], nj * xv);
    }
  }
}

// ---------------- fused dual-gate GEMM + GRU epilogue (fp32 WMMA) ----------
// Per wave: 16-row tile. acc[g][t] = 16x16 C fragment of gate g, N-tile t.
// K = 64 accumulated with V_WMMA_F32_16X16X4_F32 (16 k-steps).
// B fragments stored in LDS as packed K-pairs: Wp[g][k/2][n] = {W[k][n], W[k+1][n]}
// so each fragment is a single aligned ds_load_b64 into an even VGPR pair.

__global__ void __launch_bounds__(256)
gru_gemm_kernel(const float* __restrict__ x,
                const float* __restrict__ Tx,
                const float* __restrict__ Wx0,   // [3,32,64]
                const float* __restrict__ Wx1,   // [3,32,64]
                const float* __restrict__ bx,    // [3,64]
                const float* __restrict__ bh,    // [3,64]
                const float* __restrict__ Wlin,  // [64]
                const float* __restrict__ blin,  // [1]
                float* __restrict__ out) {
  __shared__ float Wp[2][32][64][2];  // gate {0,2}; [kpair][n][j]: W[2kp+j][n]
  __shared__ float bsum[2][64];       // bx + bh per gate
  __shared__ float wlin_s[64];

  int tid = threadIdx.x;

  // cooperative staging of weights into LDS (k<32: Wx0, k>=32: Wx1)
  for (int i = tid; i < 2 * 64 * 64; i += 256) {
    int g   = i >> 12;           // 0 -> gate0, 1 -> gate2
    int rem = i & 4095;
    int k   = rem >> 6;
    int n   = rem & 63;
    int g0  = g * 2;             // gates 0 and 2
    float v = (k < 32) ? Wx0[g0 * 2048 + k * 64 + n]
                       : Wx1[g0 * 2048 + (k - 32) * 64 + n];
    Wp[g][k >> 1][n][k & 1] = v;
  }
  for (int i = tid; i < 128; i += 256) {
    int g = i >> 6, n = i & 63, g0 = g * 2;
    bsum[g][n] = bx[g0 * 64 + n] + bh[g0 * 64 + n];
  }
  if (tid < 64) wlin_s[tid] = Wlin[tid];
  __syncthreads();

  int lane = tid & 31;
  int wave = tid >> 5;
  int tile = blockIdx.x * TILES_PER_BLOCK + wave;
  if (tile >= N_TILES) return;

  int half = lane >> 4;          // 0: K pair {0,1} / rows M 0..7 ; 1: {2,3} / M 8..15
  int mlow = lane & 15;
  int row0 = tile * 16;

  v8f acc[2][4];
  #pragma unroll
  for (int g = 0; g < 2; ++g)
    #pragma unroll
    for (int t = 0; t < 4; ++t)
      acc[g][t] = v8f{};

  const float* xr = x  + (size_t)(row0 + mlow) * 32;
  const float* tr = Tx + (size_t)(row0 + mlow) * 32;

  #pragma unroll
  for (int s = 0; s < 16; ++s) {
    int kb = 4 * s + 2 * half;   // A fragment: lane holds X2[row][kb], X2[row][kb+1]
    int kp = 2 * s + half;       // packed K-pair index
    const float* src = (kb < 32) ? (xr + kb) : (tr + (kb - 32));
    v2f a = *(const v2f*)src;
    #pragma unroll
    for (int g = 0; g < 2; ++g) {
      #pragma unroll
      for (int t = 0; t < 4; ++t) {
        v2f b = *(const v2f*)&Wp[g][kp][16 * t + mlow][0];  // one ds_load_b64
        acc[g][t] = __builtin_amdgcn_wmma_f32_16x16x4_f32(
            /*neg_a=*/false, a, /*neg_b=*/false, b,
            /*c_mod=*/(short)0, acc[g][t],
            /*reuse_a=*/false, /*reuse_b=*/false);
      }
    }
  }

  // epilogue: H = (1 - sigmoid(g0 + b0)) * tanh(g2 + b2); p = H . Wlin
  float p[8];
  #pragma unroll
  for (int r = 0; r < 8; ++r) p[r] = 0.0f;

  #pragma unroll
  for (int t = 0; t < 4; ++t) {
    int n = 16 * t + mlow;
    float b0 = bsum[0][n];
    float b2 = bsum[1][n];
    float wl = wlin_s[n];
    #pragma unroll
    for (int r = 0; r < 8; ++r) {
      float z  = 1.0f / (1.0f + __expf(-(acc[0][t][r] + b0)));
      float ht = tanhf(acc[1][t][r] + b2);
      p[r] += (1.0f - z) * ht * wl;
    }
  }

  // reduce across the 16 lanes of each N-group (C fragment row M = r + 8*half)
  #pragma unroll
  for (int r = 0; r < 8; ++r) {
    #pragma unroll
    for (int off = 1; off < 16; off <<= 1)
      p[r] += __shfl_xor(p[r], off, 16);
  }

  if (mlow == 0) {
    float bl = blin[0];
    #pragma unroll
    for (int r = 0; r < 8; ++r)
      out[row0 + 8 * half + r] = p[r] + bl;
  }
}

// ---------------------------------------------------------------------------

extern "C" void kernel_launch(void* const* d_in, const int* in_sizes, int n_in,
                              void* d_out, int out_size, void* d_ws, size_t ws_size,
                              hipStream_t stream) {
  const float* x    = (const float*)d_in[0];   // [N,32]
  const float* ew   = (const float*)d_in[1];   // [E]
  const float* Wx0  = (const float*)d_in[2];   // [3,32,64]
  const float* Wx1  = (const float*)d_in[3];   // [3,32,64]
  const float* bx   = (const float*)d_in[4];   // [3,64]
  const float* bh   = (const float*)d_in[7];   // [3,64]
  const float* Wlin = (const float*)d_in[8];   // [64,1]
  const float* blin = (const float*)d_in[9];   // [1]
  const int*   ei   = (const int*)d_in[10];    // [2,E]
  const int* row = ei;
  const int* col = ei + N_EDGES;
  float* out = (float*)d_out;

  // workspace layout: deg[N] | Tx[N*32] | disq[N] | nrm[E]
  float* deg  = (float*)d_ws;
  float* Tx   = deg + N_NODES;
  float* disq = Tx + (size_t)N_NODES * 32;
  float* nrm  = disq + N_NODES;

  const int T = 256;
  // zero deg + Tx in one shot (they are adjacent): 33*N floats
  int nz = N_NODES * 33;
  zero_f32_kernel<<<(nz + T - 1) / T, T, 0, stream>>>(deg, nz);

  deg_kernel<<<(N_EDGES + T - 1) / T, T, 0, stream>>>(ew, row, deg, N_EDGES);
  disq_kernel<<<(N_NODES + T - 1) / T, T, 0, stream>>>(deg, disq, N_NODES);
  norm_kernel<<<(N_EDGES + T - 1) / T, T, 0, stream>>>(ew, row, col, disq, nrm, N_EDGES);

  scatter_kernel<<<1024, T, 0, stream>>>(x, row, col, nrm, Tx, N_EDGES);

  int nblocks = (N_TILES + TILES_PER_BLOCK - 1) / TILES_PER_BLOCK; // 391
  gru_gemm_kernel<<<nblocks, T, 0, stream>>>(x, Tx, Wx0, Wx1, bx, bh, Wlin, blin, out);
}